// PINNLoss_20925080666375
// MI455X (gfx1250) — compile-verified
//
#include <hip/hip_runtime.h>
#include <cstdint>
#include <cstddef>

// ---------------------------------------------------------------------------
// PINN wave-equation residual norm, MI455X (gfx1250).
// Memory-bound stencil: 128 MiB input, ~5.8 us floor at 23.3 TB/s HBM. The
// whole tensor fits in the 192 MB L2, so only the async-staged first touch
// goes to HBM; t+-1 neighbor re-reads hit L2. CDNA5 path used:
// GLOBAL_LOAD_ASYNC_TO_LDS_B128 (ASYNCcnt) to stage the reusable center
// plane into LDS. Compute is 4-wide per thread so vmem/ds traffic is issued
// as B128 ops (minimum instructions per byte).
// ---------------------------------------------------------------------------

namespace {
constexpr int BATCH   = 8;
constexpr int TDIM    = 64;
constexpr int YDIM    = 256;
constexpr int XDIM    = 256;
constexpr int TSEL    = TDIM - 2;        // residual planes: t = 1..62
constexpr int ROWS    = 32;              // y-rows per workgroup tile
constexpr int YTILES  = YDIM / ROWS;     // 8
constexpr int THREADS = 256;             // 8 x wave32
constexpr int NBLOCKS = BATCH * TSEL * YTILES;  // 3968

constexpr int XVEC    = 4;               // columns per thread (float4)
constexpr int XTHREADS = XDIM / XVEC;    // 64 threads span a row
constexpr int RGROUPS  = THREADS / XTHREADS;  // 4
constexpr int RPT      = ROWS / RGROUPS;      // 8 rows per thread

constexpr float INV_DH2 = 1.0f / (5.0f * 5.0f);
constexpr float INV_DT2 = 1.0f / (0.002f * 0.002f);
constexpr float INV_C2  = 1.0f / (2200.0f * 2200.0f);
} // namespace

// ---- CDNA5 async global->LDS helpers --------------------------------------

typedef int v4i __attribute__((ext_vector_type(4)));
typedef __attribute__((address_space(1))) v4i* global_v4i_ptr;
typedef __attribute__((address_space(3))) v4i* lds_v4i_ptr;

__device__ __forceinline__ void async_copy_b128(const float* g, float* l) {
#if defined(__gfx1250__)
#if __has_builtin(__builtin_amdgcn_global_load_async_to_lds_b128)
  __builtin_amdgcn_global_load_async_to_lds_b128(
      (global_v4i_ptr)(g),
      (lds_v4i_ptr)(l),
      /*imm offset*/ 0, /*cpol*/ 0);
#else
  unsigned lds_off =
      (unsigned)(uintptr_t)(__attribute__((address_space(3))) float*)l;
  asm volatile("global_load_async_to_lds_b128 %0, %1, off"
               :: "v"(lds_off), "v"(g)
               : "memory");
#endif
#else
  // host pass / non-gfx1250: plain copy so Sema is happy; never executed.
  l[0] = g[0]; l[1] = g[1]; l[2] = g[2]; l[3] = g[3];
#endif
}

__device__ __forceinline__ void wait_asynccnt0() {
#if defined(__gfx1250__)
#if __has_builtin(__builtin_amdgcn_s_wait_asynccnt)
  __builtin_amdgcn_s_wait_asynccnt(0);
#else
  asm volatile("s_wait_asynccnt 0" ::: "memory");
#endif
#endif
}

// ---- Stage-1 kernel: per-block partial sum of squared residuals -----------

__global__ __launch_bounds__(THREADS)
void pinn_residual_partial(const float* __restrict__ u,
                           float* __restrict__ partial) {
  __shared__ __align__(16) float tile[(ROWS + 2) * XDIM];  // ~34 KB

  const int bid   = blockIdx.x;
  const int ytile = bid & (YTILES - 1);
  const int tIdx  = (bid / YTILES) % TSEL;
  const int b     = bid / (YTILES * TSEL);
  const int t     = tIdx + 1;                 // 1..62
  const int y0    = ytile * ROWS;
  const int tid   = threadIdx.x;

  const size_t planeStride = (size_t)YDIM * XDIM;
  const float* plane   = u + ((size_t)b * TDIM + t) * planeStride;
  const float* planeTm = plane - planeStride;  // t-1 (t>=1 always valid)
  const float* planeTp = plane + planeStride;  // t+1 (t<=62 always valid)

  // Stage center-plane rows [y0-1, y0+ROWS] (34 x 256 f32) into LDS.
  // In-range rows: async B128 copies (ASYNCcnt). Padded halo rows: zeros.
  constexpr int CHUNKS_PER_ROW = XDIM / 4;              // 64 x 16B
  constexpr int CHUNKS = (ROWS + 2) * CHUNKS_PER_ROW;   // 2176
  for (int idx = tid; idx < CHUNKS; idx += THREADS) {
    const int row = idx >> 6;        // / CHUNKS_PER_ROW
    const int ch  = idx & 63;        // % CHUNKS_PER_ROW
    const int gy  = y0 - 1 + row;
    float* ldst = &tile[row * XDIM + ch * 4];
    if (gy >= 0 && gy < YDIM) {
      async_copy_b128(plane + (size_t)gy * XDIM + (size_t)ch * 4, ldst);
    } else {
      *(float4*)ldst = make_float4(0.f, 0.f, 0.f, 0.f);
    }
  }
  wait_asynccnt0();
  __syncthreads();

  // Thread owns 4 consecutive columns and 8 rows:
  //   xbase = 4*(tid % 64), rows r0..r0+7 with r0 = 8*(tid / 64).
  // Output row y0+r uses tile rows r (ym), r+1 (c), r+2 (yp).
  const int xbase = (tid & (XTHREADS - 1)) * XVEC;
  const int r0    = (tid >> 6) * RPT;
  const bool hasL = (xbase > 0);
  const bool hasR = (xbase < XDIM - XVEC);

  float4 ym = *(const float4*)&tile[(r0 + 0) * XDIM + xbase];
  float4 c  = *(const float4*)&tile[(r0 + 1) * XDIM + xbase];
  float acc = 0.f;

#pragma unroll
  for (int i = 0; i < RPT; ++i) {
    const int r    = r0 + i;
    const int rowC = (r + 1) * XDIM;   // tile row holding the center y
    const float4 yp = *(const float4*)&tile[(r + 2) * XDIM + xbase];

    // Left/right halo scalars. Raw reads stay inside the tile array
    // (rowC >= XDIM, rowC + xbase + 4 <= 33*XDIM), select picks the pad 0.
    const float xlr = tile[rowC + xbase - 1];
    const float xrr = tile[rowC + xbase + XVEC];
    const float xl  = hasL ? xlr : 0.f;
    const float xr  = hasR ? xrr : 0.f;

    // t+-1 planes: one B128 load each, coalesced, L2-resident re-reads.
    const size_t off = (size_t)(y0 + r) * XDIM + xbase;
    const float4 tm = *(const float4*)&planeTm[off];
    const float4 tp = *(const float4*)&planeTp[off];

    // x-shifted neighbors from registers.
    const float xm0 = xl,  xm1 = c.x, xm2 = c.y, xm3 = c.z;
    const float xp0 = c.y, xp1 = c.z, xp2 = c.w, xp3 = xr;

    {
      const float tc = 2.0f * c.x;
      const float res = (xm0 - tc + xp0) * INV_DH2 + (ym.x - tc + yp.x) * INV_DH2
                      - ((tm.x - tc + tp.x) * INV_DT2) * INV_C2;
      acc = fmaf(res, res, acc);
    }
    {
      const float tc = 2.0f * c.y;
      const float res = (xm1 - tc + xp1) * INV_DH2 + (ym.y - tc + yp.y) * INV_DH2
                      - ((tm.y - tc + tp.y) * INV_DT2) * INV_C2;
      acc = fmaf(res, res, acc);
    }
    {
      const float tc = 2.0f * c.z;
      const float res = (xm2 - tc + xp2) * INV_DH2 + (ym.z - tc + yp.z) * INV_DH2
                      - ((tm.z - tc + tp.z) * INV_DT2) * INV_C2;
      acc = fmaf(res, res, acc);
    }
    {
      const float tc = 2.0f * c.w;
      const float res = (xm3 - tc + xp3) * INV_DH2 + (ym.w - tc + yp.w) * INV_DH2
                      - ((tm.w - tc + tp.w) * INV_DT2) * INV_C2;
      acc = fmaf(res, res, acc);
    }

    ym = c;
    c  = yp;
  }

  // Block tree-reduction (reuse tile LDS).
  __syncthreads();
  tile[tid] = acc;
  __syncthreads();
  for (int s = THREADS >> 1; s > 0; s >>= 1) {
    if (tid < s) tile[tid] += tile[tid + s];
    __syncthreads();
  }
  if (tid == 0) partial[bid] = tile[0];
}

// ---- Stage-2 kernel: deterministic reduction of block partials + sqrt -----

__global__ __launch_bounds__(256)
void pinn_finalize(const float* __restrict__ partial, int n,
                   float* __restrict__ out) {
  __shared__ float red[256];
  float s = 0.f;
  for (int i = threadIdx.x; i < n; i += 256) s += partial[i];
  red[threadIdx.x] = s;
  __syncthreads();
  for (int k = 128; k > 0; k >>= 1) {
    if (threadIdx.x < k) red[threadIdx.x] += red[threadIdx.x + k];
    __syncthreads();
  }
  if (threadIdx.x == 0) out[0] = sqrtf(red[0]);
}

// ---- Entry point ----------------------------------------------------------

extern "C" void kernel_launch(void* const* d_in, const int* in_sizes, int n_in,
                              void* d_out, int out_size, void* d_ws,
                              size_t ws_size, hipStream_t stream) {
  (void)in_sizes; (void)n_in; (void)out_size; (void)ws_size;
  const float* u = (const float*)d_in[0];       // (8,1,64,256,256) f32
  float* out     = (float*)d_out;               // scalar f32
  float* partial = (float*)d_ws;                // NBLOCKS floats (~15.5 KB)

  pinn_residual_partial<<<NBLOCKS, THREADS, 0, stream>>>(u, partial);
  pinn_finalize<<<1, 256, 0, stream>>>(partial, NBLOCKS, out);
}